// LCNSpiking_4698694222618
// MI455X (gfx1250) — compile-verified
//
#include <hip/hip_runtime.h>

#define LCN_K     25
#define BATCH     256
#define D0        14400
#define IN_STRIDE (10 * 14400)   // per-batch stride in input
#define LAST_OFF  (9 * 14400)    // offset of last timestep
#define FC_K      450
#define FC_KPAD   452

typedef float v2f __attribute__((ext_vector_type(2)));
typedef float v8f __attribute__((ext_vector_type(8)));

// ---------------------------------------------------------------------------
// 1) Transpose last timestep: input(B, 10, 14400) -> xt(14400, 256)
//    LDS 32x32 tile (33 pitch: no bank conflicts), coalesced both ways.
// ---------------------------------------------------------------------------
__global__ __launch_bounds__(256) void k_transpose(const float* __restrict__ in,
                                                   float* __restrict__ out) {
  __shared__ float tile[32][33];
  const int tx = threadIdx.x;            // 0..31
  const int ty = threadIdx.y;            // 0..7
  const int bBase = blockIdx.x * 32;     // batch tile
  const int dBase = blockIdx.y * 32;     // feature tile
#pragma unroll
  for (int i = 0; i < 4; ++i) {
    const int b = bBase + ty + i * 8;
    tile[ty + i * 8][tx] = in[(size_t)b * IN_STRIDE + LAST_OFF + dBase + tx];
  }
  __syncthreads();
#pragma unroll
  for (int i = 0; i < 4; ++i) {
    const int d = dBase + ty + i * 8;
    out[(size_t)d * BATCH + bBase + tx] = tile[tx][ty + i * 8];
  }
}

// ---------------------------------------------------------------------------
// 2) LCN layer: xout[d, b] = sum_k xin[knn[d,k], b] * w[d,k] + bias[d]
//    threadIdx.x = batch lane (coalesced gathers), knn/w via scalar loads.
// ---------------------------------------------------------------------------
__global__ __launch_bounds__(256) void k_lcn(const float* __restrict__ xin,
                                             float* __restrict__ xout,
                                             const int* __restrict__ knn,
                                             const float* __restrict__ w,
                                             const float* __restrict__ bias,
                                             int dOut) {
  const int b  = threadIdx.x;
  const int d0 = blockIdx.x * 8;
  const int dE = (d0 + 8 < dOut) ? (d0 + 8) : dOut;
  for (int d = d0; d < dE; ++d) {
    const int*   kk = knn + d * LCN_K;   // wave-uniform -> s_load
    const float* ww = w   + d * LCN_K;   // wave-uniform -> s_load
    float acc = 0.0f;
#pragma unroll
    for (int k = 0; k < LCN_K; ++k) {
      acc = fmaf(xin[(size_t)kk[k] * BATCH + b], ww[k], acc);
    }
    xout[(size_t)d * BATCH + b] = acc + bias[d];
  }
}

// ---------------------------------------------------------------------------
// 3) FC via V_WMMA_F32_16X16X4_F32: out(256,2) = xt^T(256,450) @ fcw^T(450,2)
//    One wave per 16-batch tile (grid=16). N padded 2->16, K padded 450->452.
//    A lane layout: m = lane&15, k = kbase + (lane>>4)*2 + {0,1}
//    B lane layout: n = lane&15, k = kbase + (lane>>4)*2 + {0,1}
//    D lane layout: n = lane&15, row m = v + 8*(lane>>4) for VGPR v in 0..7
// ---------------------------------------------------------------------------
__global__ __launch_bounds__(32) void k_fc_wmma(const float* __restrict__ xt,   // (450,256)
                                                const float* __restrict__ fcw,  // (2,450)
                                                const float* __restrict__ fcb,  // (2,)
                                                float* __restrict__ out) {      // (256,2)
  const int lane  = threadIdx.x;
  const int mBase = blockIdx.x * 16;
  const int mIdx  = lane & 15;
  const int kHalf = (lane >> 4) << 1;        // 0 or 2
  const int n     = mIdx;
  const int nc    = (n < 2) ? n : 0;         // clamped row of fcw (select, no branch)

  v8f acc = {};
  for (int kb = 0; kb < 448; kb += 4) {
    const int k0 = kb + kHalf;
    v2f a, bb;
    a.x = xt[(size_t)(k0    ) * BATCH + mBase + mIdx];
    a.y = xt[(size_t)(k0 + 1) * BATCH + mBase + mIdx];
    const float w0 = fcw[nc * FC_K + k0];
    const float w1 = fcw[nc * FC_K + k0 + 1];
    bb.x = (n < 2) ? w0 : 0.0f;
    bb.y = (n < 2) ? w1 : 0.0f;
    acc = __builtin_amdgcn_wmma_f32_16x16x4_f32(false, a, false, bb,
                                                (short)0, acc, false, false);
  }
  {
    // tail: k = 448,449 valid; k = 450,451 are zero padding
    const int  k0  = 448 + kHalf;
    const bool in0 = (k0     < FC_K);
    const bool in1 = (k0 + 1 < FC_K);
    const int  k0c = in0 ? k0       : 0;
    const int  k1c = in1 ? (k0 + 1) : 0;
    const float a0 = xt[(size_t)k0c * BATCH + mBase + mIdx];
    const float a1 = xt[(size_t)k1c * BATCH + mBase + mIdx];
    const float w0 = fcw[nc * FC_K + k0c];
    const float w1 = fcw[nc * FC_K + k1c];
    v2f a, bb;
    a.x  = in0 ? a0 : 0.0f;
    a.y  = in1 ? a1 : 0.0f;
    bb.x = (in0 && (n < 2)) ? w0 : 0.0f;
    bb.y = (in1 && (n < 2)) ? w1 : 0.0f;
    acc = __builtin_amdgcn_wmma_f32_16x16x4_f32(false, a, false, bb,
                                                (short)0, acc, false, false);
  }
  if (n < 2) {                               // divergence only after all WMMAs
    const float bias = fcb[n];
    const int   mTop = (lane >> 4) << 3;     // 0 or 8
#pragma unroll
    for (int v = 0; v < 8; ++v) {
      out[(mBase + mTop + v) * 2 + n] = acc[v] + bias;
    }
  }
}

// ---------------------------------------------------------------------------
// Launch
// ---------------------------------------------------------------------------
extern "C" void kernel_launch(void* const* d_in, const int* in_sizes, int n_in,
                              void* d_out, int out_size, void* d_ws, size_t ws_size,
                              hipStream_t stream) {
  (void)in_sizes; (void)n_in; (void)out_size; (void)ws_size;

  const float* input = (const float*)d_in[0];
  const int*   knn[5] = {(const int*)d_in[1],  (const int*)d_in[4],
                         (const int*)d_in[7],  (const int*)d_in[10],
                         (const int*)d_in[13]};
  const float* w[5]   = {(const float*)d_in[2],  (const float*)d_in[5],
                         (const float*)d_in[8],  (const float*)d_in[11],
                         (const float*)d_in[14]};
  const float* bs[5]  = {(const float*)d_in[3],  (const float*)d_in[6],
                         (const float*)d_in[9],  (const float*)d_in[12],
                         (const float*)d_in[15]};
  const float* fcw = (const float*)d_in[16];
  const float* fcb = (const float*)d_in[17];
  float*       out = (float*)d_out;

  float* bufA = (float*)d_ws;                       // 14400*256 floats
  float* bufB = bufA + (size_t)D0 * BATCH;          //  7200*256 floats

  // x^T = input[:, -1, :]^T  -> (14400, 256) feature-major
  k_transpose<<<dim3(BATCH / 32, D0 / 32), dim3(32, 8), 0, stream>>>(input, bufA);

  static const int dims[6] = {14400, 7200, 3600, 1800, 900, 450};
  float* cur = bufA;
  float* nxt = bufB;
  for (int i = 0; i < 5; ++i) {
    const int dOut = dims[i + 1];
    const int grid = (dOut + 7) / 8;
    k_lcn<<<grid, 256, 0, stream>>>(cur, nxt, knn[i], w[i], bs[i], dOut);
    float* t = cur; cur = nxt; nxt = t;
  }
  // cur == bufB, holds (450, 256)
  k_fc_wmma<<<16, 32, 0, stream>>>(cur, fcw, fcb, out);
}